// Heirachical_Loss_68521908241003
// MI455X (gfx1250) — compile-verified
//
#include <hip/hip_runtime.h>
#include <hip/hip_bf16.h>
#include <math.h>

// Problem constants (from reference): B=1048576, L=81, G=9
#define LEAF 81
#define GRP 9
#define ROWS 256                  // rows per block == threads per block
#define TILE_DW (ROWS * LEAF)     // 20736 dwords = 82944 bytes of LDS
#define TILE_V4 (TILE_DW / 4)     // 5184 float4 chunks = 256*20 + 64
#define FULL_STEPS 20             // full 256-lane float4 rounds
#define TAIL_V4 (TILE_V4 - FULL_STEPS * ROWS)   // 64 remaining float4s

#if defined(__gfx1250__) && __has_builtin(__builtin_amdgcn_global_load_async_to_lds_b128) && __has_builtin(__builtin_amdgcn_s_wait_asynccnt)
#define USE_ASYNC_LDS 1
#else
#define USE_ASYNC_LDS 0
#endif

#if USE_ASYNC_LDS
typedef int v4i __attribute__((ext_vector_type(4)));
typedef __attribute__((address_space(1))) v4i* g_v4i_ptr;   // global
typedef __attribute__((address_space(3))) v4i* l_v4i_ptr;   // LDS

// Compile-time unrolled async-copy: offset is a template parameter, i.e. a
// genuine integer constant expression as the builtin requires. The immediate
// offset applies to BOTH the global and LDS addresses (ISA: async LDS loads),
// which advance in lockstep here — so one base VGPR pair serves all steps.
template <int K>
struct AsyncSteps {
    static __device__ __forceinline__ void run(g_v4i_ptr gp, l_v4i_ptr lp) {
        AsyncSteps<K - 1>::run(gp, lp);
        __builtin_amdgcn_global_load_async_to_lds_b128(gp, lp, (K - 1) * ROWS * 16, 0);
    }
};
template <>
struct AsyncSteps<0> {
    static __device__ __forceinline__ void run(g_v4i_ptr, l_v4i_ptr) {}
};
#endif

__global__ __launch_bounds__(ROWS) void hloss_main_kernel(
    const float* __restrict__ outputs,   // [B, 81]
    const int*   __restrict__ target,    // [B]
    float* __restrict__ out,             // [1 + B + 1]: loss, pred(as float), dist
    float* __restrict__ lossPart,        // [nblk]
    int*   __restrict__ distPart)        // [nblk]
{
    __shared__ __align__(16) float smem[TILE_DW];

    const int tid = threadIdx.x;
    const int blk = blockIdx.x;
    const float* __restrict__ gsrc = outputs + (size_t)blk * TILE_DW;

    // ---- Stage 256x81 tile into LDS, fully coalesced 16B chunks ----
#if USE_ASYNC_LDS
    {
        g_v4i_ptr gp = (g_v4i_ptr)(gsrc + 4 * tid);
        l_v4i_ptr lp = (l_v4i_ptr)(smem + 4 * tid);
        AsyncSteps<FULL_STEPS>::run(gp, lp);
        if (tid < TAIL_V4)
            __builtin_amdgcn_global_load_async_to_lds_b128(
                (g_v4i_ptr)(gsrc + 4 * (FULL_STEPS * ROWS + tid)),
                (l_v4i_ptr)(smem + 4 * (FULL_STEPS * ROWS + tid)), 0, 0);
        __builtin_amdgcn_s_wait_asynccnt(0);
    }
#else
    for (int i = tid; i < TILE_V4; i += ROWS)
        *(float4*)(smem + 4 * i) = *(const float4*)(gsrc + 4 * i);
#endif
    __syncthreads();

    // ---- Each thread owns one row in LDS (bank-conflict-free: 81 mod 64 = 17, odd) ----
    const int   row = blk * ROWS + tid;
    const float* x  = smem + tid * LEAF;

    // Pass 1: row max
    float m = x[0];
#pragma unroll
    for (int i = 1; i < LEAF; ++i) m = fmaxf(m, x[i]);

    // Pass 2: exp, group sums, total sum
    float gsum[GRP];
    float sumExp = 0.f;
#pragma unroll
    for (int g = 0; g < GRP; ++g) {
        float s = 0.f;
#pragma unroll
        for (int j = 0; j < GRP; ++j) s += __expf(x[g * GRP + j] - m);
        gsum[g] = s;
        sumExp += s;
    }

    const int tgt   = target[row];
    const int par   = tgt / GRP;
    const float el  = __expf(x[tgt] - m);
    // -log(win) = log(sumExp) - log(0.5*(gsum[par] + e_leaf))   (division-free)
    const float lossB = __logf(sumExp) - __logf(0.5f * (gsum[par] + el));

    // Greedy top-down argmax: group by exp-sum, leaf by raw logit (monotone in exp)
    int   bg = 0;
    float bv = gsum[0];
#pragma unroll
    for (int g = 1; g < GRP; ++g)
        if (gsum[g] > bv) { bv = gsum[g]; bg = g; }

    int   wi = 0;
    float wv = x[bg * GRP];
#pragma unroll
    for (int j = 1; j < GRP; ++j) {
        float v = x[bg * GRP + j];
        if (v > wv) { wv = v; wi = j; }
    }
    const int pred = bg * GRP + wi;
    out[1 + row] = (float)pred;

    const int dist = (pred == tgt) ? 0 : ((bg == par) ? 1 : 2);

    // ---- Deterministic block reduction (reuse tile LDS after barrier) ----
    __syncthreads();
    float* rf = smem;                 // 256 floats
    int*   ri = (int*)(smem + ROWS);  // 256 ints
    rf[tid] = lossB;
    ri[tid] = dist;
    __syncthreads();
#pragma unroll
    for (int s = ROWS / 2; s > 0; s >>= 1) {
        if (tid < s) { rf[tid] += rf[tid + s]; ri[tid] += ri[tid + s]; }
        __syncthreads();
    }
    if (tid == 0) { lossPart[blk] = rf[0]; distPart[blk] = ri[0]; }
}

__global__ __launch_bounds__(256) void hloss_final_kernel(
    const float* __restrict__ lossPart,
    const int*   __restrict__ distPart,
    float* __restrict__ out,
    int nblk, int btotal)
{
    __shared__ float sf[256];
    __shared__ int   si[256];
    const int tid = threadIdx.x;
    float s = 0.f;
    int   d = 0;
    for (int i = tid; i < nblk; i += 256) { s += lossPart[i]; d += distPart[i]; }
    sf[tid] = s;
    si[tid] = d;
    __syncthreads();
#pragma unroll
    for (int k = 128; k > 0; k >>= 1) {
        if (tid < k) { sf[tid] += sf[tid + k]; si[tid] += si[tid + k]; }
        __syncthreads();
    }
    if (tid == 0) {
        out[0]          = sf[0] / (float)btotal;   // mean loss
        out[1 + btotal] = (float)si[0];            // total tree distance
    }
}

extern "C" void kernel_launch(void* const* d_in, const int* in_sizes, int n_in,
                              void* d_out, int out_size, void* d_ws, size_t ws_size,
                              hipStream_t stream) {
    const float* outputs = (const float*)d_in[0];
    const int*   target  = (const int*)d_in[1];
    float*       out     = (float*)d_out;

    const int btotal = in_sizes[1];      // 1048576 (rows); divisible by ROWS
    const int nblk   = btotal / ROWS;    // 4096

    float* lossPart = (float*)d_ws;
    int*   distPart = (int*)((char*)d_ws + (size_t)nblk * sizeof(float));

    hloss_main_kernel<<<nblk, ROWS, 0, stream>>>(outputs, target, out, lossPart, distPart);
    hloss_final_kernel<<<1, 256, 0, stream>>>(lossPart, distPart, out, nblk, btotal);
}